// ChamferDistanceLoss_18863496364104
// MI455X (gfx1250) — compile-verified
//
#include <hip/hip_runtime.h>

typedef __attribute__((ext_vector_type(2))) float v2f;
typedef __attribute__((ext_vector_type(8))) float v8f;

#define CD_B 8
#define BLOCK_THREADS 256
#define WAVES_PER_BLOCK (BLOCK_THREADS / 32)

// Unified chamfer pass.
//
// Each wave owns 16 "row" points R and computes, for each of them,
//   min over all "loop" points Q of dist(R, Q),
// then the sum of those 16 mins -> partial[job].
//
// Family 0: rows = pred,   loop = target  (min_p2t contribution)
// Family 1: rows = target, loop = pred    (min_t2p contribution)
//
// Distance via rank-4 WMMA:  A[m,:] = (-2Rx, -2Ry, -2Rz, 1),
//                            B[:,n] = (Qx, Qy, Qz, |Q|^2)
//   => D = -2 R.Q + |Q|^2 ;  |R|^2 added after the row-min reduction.
//
// The loop-side operand is cooperatively packed into LDS once per block in the
// exact per-lane WMMA B layout (lanes 0-15: (x,y) = K0,K1; lanes 16-31:
// (z,|q|^2) = K2,K3), so the hot loop is ds_load_b64 + v_wmma + 8 fmin.
__global__ void __launch_bounds__(BLOCK_THREADS)
chamfer_main_kernel(const float* __restrict__ pred, const float* __restrict__ target,
                    float* __restrict__ partial, int N, int M) {
    extern __shared__ float ldsraw[];
    float2* pack = (float2*)ldsraw;  // [loopTiles][32] float2, 16B-aligned

    const int rowJobs  = CD_B * (N >> 4);
    const int blockJob = blockIdx.x * WAVES_PER_BLOCK;  // 8 jobs, same family+batch
    const bool fam1    = blockJob >= rowJobs;

    const int rowLen   = fam1 ? M : N;   // points on the row side
    const int loopLen  = fam1 ? N : M;   // points on the loop side
    const int tilesRow = rowLen >> 4;

    const int jobLocal = fam1 ? (blockJob - rowJobs) : blockJob;
    const int b        = jobLocal / tilesRow;  // uniform across the block

    const float* rowPts  = (fam1 ? target : pred) + (size_t)b * rowLen * 3;
    const float* loopPts = (fam1 ? pred : target) + (size_t)b * loopLen * 3;

    // ---- Cooperative pack of the loop-side batch into LDS (WMMA B layout) ----
    for (int j = threadIdx.x; j < loopLen; j += BLOCK_THREADS) {
        const float* q = loopPts + (size_t)j * 3;
        const float x = q[0], y = q[1], z = q[2];
        const float sq = x * x + y * y + z * z;
        const int t = j >> 4, i = j & 15;
        pack[t * 32 + i]      = make_float2(x, y);   // lanes 0-15: K0,K1
        pack[t * 32 + i + 16] = make_float2(z, sq);  // lanes 16-31: K2,K3
    }
    __syncthreads();

    // ---- Per-wave setup ----
    const int waveJob = blockJob + (threadIdx.x >> 5);
    const int jt      = (fam1 ? (waveJob - rowJobs) : waveJob) % tilesRow;
    const int lane    = threadIdx.x & 31;
    const int half    = lane >> 4;
    const int i16     = lane & 15;

    const float* r = rowPts + ((size_t)jt * 16 + i16) * 3;
    const float rx = r[0], ry = r[1], rz = r[2];
    const float rsq = rx * rx + ry * ry + rz * rz;

    // A (16x4 f32): lanes 0-15 hold (K0,K1), lanes 16-31 hold (K2,K3).
    v2f a;
    a.x = half ? (-2.0f * rz) : (-2.0f * rx);
    a.y = half ? 1.0f         : (-2.0f * ry);

    // |R|^2 for D-row m = v + 8*half (needed after the min reduction).
    float rsqv[8];
#pragma unroll
    for (int v = 0; v < 8; ++v)
        rsqv[v] = __shfl(rsq, v + 8 * half, 32);

    v8f acc;
#pragma unroll
    for (int v = 0; v < 8; ++v) acc[v] = 3.0e38f;

    // ---- Hot loop: one WMMA tile per iteration ----
    const float2* lp = pack + lane;
    const int loopTiles = loopLen >> 4;
#pragma unroll 4
    for (int t = 0; t < loopTiles; ++t) {
        const float2 e = lp[t * 32];
        v2f bm;
        bm.x = e.x;
        bm.y = e.y;

        v8f c = {};
        v8f d = __builtin_amdgcn_wmma_f32_16x16x4_f32(
            false, a, false, bm, (short)0, c, false, false);

#pragma unroll
        for (int v = 0; v < 8; ++v) acc[v] = fminf(acc[v], d[v]);
    }

    // ---- Horizontal min per D-row (within each 16-lane group), add |R|^2, sum ----
    float s = 0.0f;
#pragma unroll
    for (int v = 0; v < 8; ++v) {
        float m = acc[v];
        m = fminf(m, __shfl_xor(m, 1, 32));
        m = fminf(m, __shfl_xor(m, 2, 32));
        m = fminf(m, __shfl_xor(m, 4, 32));
        m = fminf(m, __shfl_xor(m, 8, 32));
        s += m + rsqv[v];
    }
    s += __shfl_xor(s, 16, 32);  // rows 0-7 half + rows 8-15 half
    if (lane == 0) partial[waveJob] = s;
}

// Single-wave deterministic final reduction.
__global__ void chamfer_reduce_kernel(const float* __restrict__ partial,
                                      float* __restrict__ out,
                                      int rowJobs, int colJobs,
                                      float invBN, float invBM) {
    const int lane = threadIdx.x;
    float sr = 0.0f, sc = 0.0f;
    for (int i = lane; i < rowJobs; i += 32) sr += partial[i];
    for (int i = lane; i < colJobs; i += 32) sc += partial[rowJobs + i];
#pragma unroll
    for (int m = 16; m >= 1; m >>= 1) {
        sr += __shfl_xor(sr, m, 32);
        sc += __shfl_xor(sc, m, 32);
    }
    if (lane == 0) out[0] = sr * invBN + sc * invBM;
}

extern "C" void kernel_launch(void* const* d_in, const int* in_sizes, int n_in,
                              void* d_out, int out_size, void* d_ws, size_t ws_size,
                              hipStream_t stream) {
    const float* pred   = (const float*)d_in[0];
    const float* target = (const float*)d_in[1];
    float* out = (float*)d_out;
    float* partial = (float*)d_ws;

    const int N = in_sizes[0] / (CD_B * 3);  // 4096
    const int M = in_sizes[1] / (CD_B * 3);  // 4096

    const int rowJobs = CD_B * (N >> 4);     // 2048 pred-side waves
    const int colJobs = CD_B * (M >> 4);     // 2048 target-side waves

    const int totalBlocks = (rowJobs + colJobs) / WAVES_PER_BLOCK;  // 512
    const int maxLoop = (N > M) ? N : M;
    const size_t ldsBytes = (size_t)maxLoop * 16;  // 64 KB: loopTiles*32 float2

    chamfer_main_kernel<<<totalBlocks, BLOCK_THREADS, ldsBytes, stream>>>(
        pred, target, partial, N, M);
    chamfer_reduce_kernel<<<1, 32, 0, stream>>>(
        partial, out, rowJobs, colJobs,
        1.0f / ((float)CD_B * (float)N), 1.0f / ((float)CD_B * (float)M));
}